// MultiPositionTransfer_62723702390979
// MI455X (gfx1250) — compile-verified
//
#include <hip/hip_runtime.h>

#define MULTI_INFER 8
#define D 128
#define NROWS 16384           // T*B = 512*32
#define NBUCKET 9             // positions clipped to 0..8
#define TILE_M 16
#define MAX_TILES (NROWS / TILE_M + NBUCKET)   // 1033: worst-case bucket padding

typedef __attribute__((ext_vector_type(2))) float v2f;
typedef __attribute__((ext_vector_type(8))) float v8f;

// ---------------- bucketing kernels ----------------

__global__ void init_ws(int* counts) {
    int t = threadIdx.x;
    if (t < 48) counts[t] = 0;   // counts[16] + base[16] + cursor[16]
}

__global__ void count_pos(const long long* __restrict__ pos, int* __restrict__ counts) {
    int i = blockIdx.x * blockDim.x + threadIdx.x;
    if (i >= NROWS) return;
    long long v = pos[i];
    int p = (v < MULTI_INFER) ? (int)v : MULTI_INFER;
    atomicAdd(&counts[p], 1);
}

__global__ void scan_counts(const int* __restrict__ counts, int* __restrict__ base) {
    if (threadIdx.x == 0) {
        int acc = 0;
        for (int q = 0; q < NBUCKET; ++q) { base[q] = acc; acc += counts[q]; }
    }
}

__global__ void fill_lists(const long long* __restrict__ pos, const int* __restrict__ base,
                           int* __restrict__ cursor, int* __restrict__ list) {
    int i = blockIdx.x * blockDim.x + threadIdx.x;
    if (i >= NROWS) return;
    long long v = pos[i];
    int p = (v < MULTI_INFER) ? (int)v : MULTI_INFER;
    int r = atomicAdd(&cursor[p], 1);
    list[base[p] + r] = i;      // scatter-by-index => deterministic final output
}

// ---------------- WMMA GEMM kernel ----------------
// One block = one 16-row tile of a bucket. 8 waves; wave w computes the
// 16x16 output tile for columns [16w, 16w+16) as X(16x128) * M(128x128)
// via 32 V_WMMA_F32_16X16X4_F32, split into two independent accumulator
// chains (K 0..63 and 64..127) to break the back-to-back WMMA RAW chain.

__global__ __launch_bounds__(256) void
transfer_wmma(const float* __restrict__ Xg, const float* __restrict__ table,
              const int* __restrict__ counts, const int* __restrict__ base,
              const int* __restrict__ list, float* __restrict__ out)
{
    __shared__ float Xs[TILE_M][D + 4];   // stride 132 floats: b64 reads hit all 64 banks
    __shared__ int   rowIds[TILE_M];

    int tid = threadIdx.x;

    // Uniform tile -> (bucket, offset) assignment via scalar loads of counts/base.
    int tile = blockIdx.x;
    int acc = 0, p = -1, off = 0, cnt = 0, bs = 0;
    for (int q = 0; q < NBUCKET; ++q) {
        int c  = counts[q];
        int nt = (c + TILE_M - 1) >> 4;
        if (p < 0 && tile < acc + nt) {
            p = q; off = (tile - acc) * TILE_M; cnt = c; bs = base[q];
        }
        acc += nt;
    }
    if (p < 0) return;                    // past last tile (uniform exit)

    // Gather the 16 row indices for this tile (clamp padding rows to a valid row
    // so loads stay in-bounds; stores are masked by `cnt` below).
    if (tid < TILE_M) {
        int lr = off + tid;
        if (lr >= cnt) lr = cnt - 1;
        rowIds[tid] = list[bs + lr];
    }
    __syncthreads();

    // Stage X tile: 256 threads x 8 floats = 16x128.
    {
        int lrow = tid >> 4;
        int c0   = (tid & 15) * 8;
        const float* src = Xg + (size_t)rowIds[lrow] * D + c0;
        float4 u0 = *(const float4*)(src);
        float4 u1 = *(const float4*)(src + 4);
        *(float4*)&Xs[lrow][c0]     = u0;
        *(float4*)&Xs[lrow][c0 + 4] = u1;
    }
    __syncthreads();

    int wave = tid >> 5;
    int lane = tid & 31;
    int hsel = lane >> 4;      // 0: lanes 0-15 (K pair k0,k0+1), 1: lanes 16-31 (k0+2,k0+3)
    int lidx = lane & 15;      // M index for A, N index for B
    int n0   = wave * 16;
    const float* M = table + (size_t)p * (D * D);

    v8f c0 = {};
    v8f c1 = {};
#pragma unroll 4
    for (int k0 = 0; k0 < D / 2; k0 += 4) {
        // ---- chain 0: K block [k0, k0+4) ----
        {
            int k = k0 + 2 * hsel;
            v2f a;                       // A[M=lidx][k..k+1]  (ISA 16x4 f32 layout)
            a.x = Xs[lidx][k];
            a.y = Xs[lidx][k + 1];
            const float* mp = M + (size_t)k * D + n0 + lidx;
            v2f b;                       // B[K=k..k+1][N=n0+lidx] (4x16, lanes carry N)
            b.x = mp[0];
            b.y = mp[D];
            c0 = __builtin_amdgcn_wmma_f32_16x16x4_f32(
                     false, a, false, b, (short)0, c0, false, false);
        }
        // ---- chain 1: K block [k0+64, k0+68) (independent accumulator) ----
        {
            int k = k0 + (D / 2) + 2 * hsel;
            v2f a;
            a.x = Xs[lidx][k];
            a.y = Xs[lidx][k + 1];
            const float* mp = M + (size_t)k * D + n0 + lidx;
            v2f b;
            b.x = mp[0];
            b.y = mp[D];
            c1 = __builtin_amdgcn_wmma_f32_16x16x4_f32(
                     false, a, false, b, (short)0, c1, false, false);
        }
    }

    // Combine the two chains (EXEC still all-ones; plain VALU adds).
    v8f c;
#pragma unroll
    for (int i = 0; i < 8; ++i) c[i] = c0[i] + c1[i];

    // Hoist all row-id reads ahead of the divergent stores: one clause of
    // LDS broadcast reads + one wait, then address math with EXEC intact.
    int myRows[8];
#pragma unroll
    for (int i = 0; i < 8; ++i) myRows[i] = rowIds[i + 8 * hsel];

    // Scatter C: VGPR i holds rows M=i (lanes 0-15) / M=i+8 (lanes 16-31), N=lane&15.
    int ncol = n0 + lidx;
#pragma unroll
    for (int i = 0; i < 8; ++i) {
        if (off + i + 8 * hsel < cnt) {
            out[(size_t)myRows[i] * D + ncol] = c[i];
        }
    }
}

// ---------------- launcher ----------------

extern "C" void kernel_launch(void* const* d_in, const int* in_sizes, int n_in,
                              void* d_out, int out_size, void* d_ws, size_t ws_size,
                              hipStream_t stream) {
    (void)in_sizes; (void)n_in; (void)out_size; (void)ws_size;
    const long long* pos   = (const long long*)d_in[0];  // int64 positions [T,B]
    const float*     outs  = (const float*)d_in[1];      // [T,B,128] fp32
    const float*     table = (const float*)d_in[2];      // [9,128,128] fp32
    float*           out   = (float*)d_out;              // [T,B,128] fp32

    int* counts = (int*)d_ws;       // 16 ints
    int* base   = counts + 16;      // 16 ints
    int* cursor = counts + 32;      // 16 ints
    int* list   = counts + 48;      // 16384 ints

    init_ws<<<1, 64, 0, stream>>>(counts);
    count_pos<<<NROWS / 256, 256, 0, stream>>>(pos, counts);
    scan_counts<<<1, 32, 0, stream>>>(counts, base);
    fill_lists<<<NROWS / 256, 256, 0, stream>>>(pos, base, cursor, list);
    transfer_wmma<<<MAX_TILES, 256, 0, stream>>>(outs, table, counts, base, list, out);
}